// SimpleMultiHeadAttention_6098853560977
// MI455X (gfx1250) — compile-verified
//
#include <hip/hip_runtime.h>
#include <hip/hip_bf16.h>
#include <math.h>

// ---------------- problem constants (static per reference) ----------------
#define D_MODEL   1024
#define N_HEADS   16
#define HEAD_DIM  64
#define SEQ_S     16
#define MAX_LEN   32768
#define START_POS (MAX_LEN - SEQ_S)        // 32752 (16-aligned)
#define NCHUNK    32
#define CHUNK     (MAX_LEN / NCHUNK)       // 1024 keys per wave
#define SCALE_F   0.125f                   // 1/sqrt(64)
#define KPAD      68                       // LDS row pitch (floats): 64 + 4 pad

typedef float v2f __attribute__((ext_vector_type(2)));
typedef float v4f __attribute__((ext_vector_type(4)));
typedef float v8f __attribute__((ext_vector_type(8)));
typedef int   v4i __attribute__((ext_vector_type(4)));

// CDNA5 async global->LDS path (ASYNCcnt), with sync fallback if unavailable.
#if defined(__has_builtin)
#if __has_builtin(__builtin_amdgcn_global_load_async_to_lds_b128) && \
    __has_builtin(__builtin_amdgcn_s_wait_asynccnt)
#define USE_ASYNC_LDS 1
#endif
#endif
#ifndef USE_ASYNC_LDS
#define USE_ASYNC_LDS 0
#endif

// Builtin prototype (from clang diagnostic): (v4i AS1*, v4i AS3*, Imm i32, Imm i32)
#define CAST_GLOBAL4(p) ((__attribute__((address_space(1))) v4i*)(uintptr_t)(p))
#define CAST_LOCAL4(p)  ((__attribute__((address_space(3))) v4i*)(unsigned int)(uintptr_t)(p))

__device__ __forceinline__ v8f wmma_f32(v2f a, v2f b, v8f c) {
  // D = A(16x4 f32) * B(4x16 f32) + C(16x16 f32)
  return __builtin_amdgcn_wmma_f32_16x16x4_f32(
      /*neg_a=*/false, a, /*neg_b=*/false, b,
      /*c_mod=*/(short)0, c, /*reuse_a=*/false, /*reuse_b=*/false);
}

// ============================================================================
// gemm16: Y[16 x 1024] = X[16 x 1024] @ W^T   (nn.Linear), one wave per
// 16-wide N tile. A-frag: lane<16 -> X[row][k..k+1], lane>=16 -> X[row][k+2..3]
// B-frag: B[kk][n] = W[n0+n][k+kk] (same striping, W rows are contiguous in k).
// ============================================================================
__global__ __launch_bounds__(32) void gemm16_kernel(
    const float* __restrict__ X, const float* __restrict__ W,
    float* __restrict__ Y, float* __restrict__ Y2) {
  const int lane = threadIdx.x;
  const int nlo  = lane & 15;
  const int hi   = lane >> 4;
  const int n0   = blockIdx.x * 16;

  const float* ap = X + (size_t)nlo * D_MODEL + 2 * hi;
  const float* bp = W + (size_t)(n0 + nlo) * D_MODEL + 2 * hi;

  v8f c = {};
  #pragma unroll 4
  for (int k = 0; k < D_MODEL; k += 4) {
    __builtin_prefetch((const void*)(bp + k + 256), 0, 1);  // global_prefetch_b8
    v2f a = *(const v2f*)(ap + k);
    v2f b = *(const v2f*)(bp + k);
    c = wmma_f32(a, b, c);
  }

  const int col = n0 + nlo;
  if (Y2 != nullptr) {
    #pragma unroll
    for (int j = 0; j < 8; ++j) {
      const int row = j + 8 * hi;               // C layout: VGPR j <-> row j+8*hi
      const float v = c[j];
      Y[(size_t)row * D_MODEL + col]  = v;
      Y2[(size_t)row * D_MODEL + col] = v;
    }
  } else {
    #pragma unroll
    for (int j = 0; j < 8; ++j) {
      const int row = j + 8 * hi;
      Y[(size_t)row * D_MODEL + col] = c[j];
    }
  }
}

// ============================================================================
// attn_partial: one wave per (head, T-chunk). Online-softmax flash decode.
// K/V tiles (16 keys x 64 f32 each) staged into LDS via async-to-LDS engine,
// double buffered. Emits per-chunk partials {m[16], l[16], acc[16][64]}.
// ============================================================================

// Stage one 16-key tile of K and V (head-sliced base pointers) into LDS.
__device__ __forceinline__ void stage_tile(const float* __restrict__ gK,
                                           const float* __restrict__ gV,
                                           float* __restrict__ lK,
                                           float* __restrict__ lV, int lane) {
  #pragma unroll
  for (int it = 0; it < 8; ++it) {
    const int idx = it * 32 + lane;            // 256 16-byte chunks per tile
    const int row = idx >> 4;                  // key row 0..15
    const int c4  = (idx & 15) * 4;            // float column 0,4,...,60
    const float* gk = gK + (size_t)row * D_MODEL + c4;
    const float* gv = gV + (size_t)row * D_MODEL + c4;
    float* lk = lK + row * KPAD + c4;
    float* lv = lV + row * KPAD + c4;
#if USE_ASYNC_LDS
    __builtin_amdgcn_global_load_async_to_lds_b128(CAST_GLOBAL4(gk), CAST_LOCAL4(lk), 0, 0);
    __builtin_amdgcn_global_load_async_to_lds_b128(CAST_GLOBAL4(gv), CAST_LOCAL4(lv), 0, 0);
#else
    *(v4f*)lk = *(const v4f*)gk;
    *(v4f*)lv = *(const v4f*)gv;
#endif
  }
}

__device__ __forceinline__ void tile_bases(int t0, int h,
                                           const float* __restrict__ cacheK,
                                           const float* __restrict__ cacheV,
                                           const float* __restrict__ Knew,
                                           const float* __restrict__ Vnew,
                                           const float*& kb, const float*& vb) {
  if (t0 < START_POS) {
    kb = cacheK + (size_t)t0 * D_MODEL + h * HEAD_DIM;
    vb = cacheV + (size_t)t0 * D_MODEL + h * HEAD_DIM;
  } else {
    kb = Knew + (size_t)(t0 - START_POS) * D_MODEL + h * HEAD_DIM;
    vb = Vnew + (size_t)(t0 - START_POS) * D_MODEL + h * HEAD_DIM;
  }
}

__global__ __launch_bounds__(32) void attn_partial_kernel(
    const float* __restrict__ Qws,  const float* __restrict__ Knew,
    const float* __restrict__ Vnew, const float* __restrict__ cacheK,
    const float* __restrict__ cacheV, float* __restrict__ part) {
  __shared__ float ldsK[2][16 * KPAD];
  __shared__ float ldsV[2][16 * KPAD];
  __shared__ float pls[16 * 17];               // P tile bounce (C -> A layout)

  const int lane  = threadIdx.x;
  const int nlo   = lane & 15;
  const int hi    = lane >> 4;
  const int h     = blockIdx.x / NCHUNK;
  const int ck    = blockIdx.x % NCHUNK;
  const int tBase = ck * CHUNK;
  const int tEnd  = tBase + CHUNK;

  // Q A-fragments for all 16 k-steps (K = HEAD_DIM = 64)
  v2f aq[16];
  {
    const float* qp = Qws + (size_t)nlo * D_MODEL + h * HEAD_DIM + 2 * hi;
    #pragma unroll
    for (int kk = 0; kk < 16; ++kk) aq[kk] = *(const v2f*)(qp + 4 * kk);
  }

  v8f acc0 = {}, acc1 = {}, acc2 = {}, acc3 = {};
  float mrow[8], lrow[8];
  #pragma unroll
  for (int j = 0; j < 8; ++j) { mrow[j] = -3.0e38f; lrow[j] = 0.0f; }

  // prologue: stage first tile into buffer 0
  {
    const float *kb, *vb;
    tile_bases(tBase, h, cacheK, cacheV, Knew, Vnew, kb, vb);
    stage_tile(kb, vb, ldsK[0], ldsV[0], lane);
  }

  int buf = 0;
  for (int t0 = tBase; t0 < tEnd; t0 += 16, buf ^= 1) {
    // stage next tile into other buffer, then wait for current tile residency
    if (t0 + 16 < tEnd) {
      const float *kb, *vb;
      tile_bases(t0 + 16, h, cacheK, cacheV, Knew, Vnew, kb, vb);
      stage_tile(kb, vb, ldsK[buf ^ 1], ldsV[buf ^ 1], lane);
#if USE_ASYNC_LDS
      __builtin_amdgcn_s_wait_asynccnt(16);    // current tile's 16 ops done
#endif
    } else {
#if USE_ASYNC_LDS
      __builtin_amdgcn_s_wait_asynccnt(0);
#endif
    }

    // ---- scores S = Q @ K^T (16x16 tile), 16 wmmas over hd=64 ----
    v8f sc = {};
    {
      const float* kp = ldsK[buf] + nlo * KPAD + 2 * hi;
      #pragma unroll
      for (int kk = 0; kk < 16; ++kk) {
        v2f b = *(const v2f*)(kp + 4 * kk);    // ds_load_b64, conflict-free
        sc = wmma_f32(aq[kk], b, sc);
      }
    }

    // ---- scale + causal mask (only last tiles can be masked) ----
    const int  tg       = t0 + nlo;            // this lane's key position
    const bool maskTile = (t0 + 15 > START_POS);
    #pragma unroll
    for (int j = 0; j < 8; ++j) {
      float v = sc[j] * SCALE_F;
      const int row = j + 8 * hi;              // query row s
      if (maskTile && (tg > row + START_POS)) v = -3.0e38f;
      sc[j] = v;
    }

    // ---- online softmax row stats (reduce across 16-lane halves) ----
    float fsc[8];
    #pragma unroll
    for (int j = 0; j < 8; ++j) {
      float mx = sc[j];
      #pragma unroll
      for (int off = 1; off < 16; off <<= 1)
        mx = fmaxf(mx, __shfl_xor(mx, off));   // xor 1,2,4,8 stays in half
      const float mnew = fmaxf(mrow[j], mx);
      const float f    = __expf(mrow[j] - mnew);
      const float p    = __expf(sc[j] - mnew);
      float sum = p;
      #pragma unroll
      for (int off = 1; off < 16; off <<= 1)
        sum += __shfl_xor(sum, off);
      lrow[j] = lrow[j] * f + sum;
      mrow[j] = mnew;
      fsc[j]  = f;
      sc[j]   = p;                             // P tile, C layout
    }
    #pragma unroll
    for (int j = 0; j < 8; ++j) {
      acc0[j] *= fsc[j]; acc1[j] *= fsc[j];
      acc2[j] *= fsc[j]; acc3[j] *= fsc[j];
    }

    // ---- P: C layout -> A layout via LDS ----
    #pragma unroll
    for (int j = 0; j < 8; ++j)
      pls[(j + 8 * hi) * 17 + nlo] = sc[j];
    __syncthreads();
    v2f ap[4];
    #pragma unroll
    for (int kk = 0; kk < 4; ++kk) {
      ap[kk].x = pls[nlo * 17 + 4 * kk + 2 * hi];
      ap[kk].y = pls[nlo * 17 + 4 * kk + 2 * hi + 1];
    }
    __syncthreads();

    // ---- O += P @ V : 4 k-steps x 4 d-tiles = 16 wmmas ----
    #pragma unroll
    for (int kk = 0; kk < 4; ++kk) {
      const float* vr = ldsV[buf] + (4 * kk + 2 * hi) * KPAD + nlo;
      v2f b0, b1, b2, b3;
      b0.x = vr[0];  b0.y = vr[KPAD];
      b1.x = vr[16]; b1.y = vr[16 + KPAD];
      b2.x = vr[32]; b2.y = vr[32 + KPAD];
      b3.x = vr[48]; b3.y = vr[48 + KPAD];
      acc0 = wmma_f32(ap[kk], b0, acc0);
      acc1 = wmma_f32(ap[kk], b1, acc1);
      acc2 = wmma_f32(ap[kk], b2, acc2);
      acc3 = wmma_f32(ap[kk], b3, acc3);
    }
  }

  // ---- emit partials: {m[16], l[16], acc[16][64]} ----
  float* pb = part + (size_t)blockIdx.x * 1056;
  if (nlo == 0) {
    #pragma unroll
    for (int j = 0; j < 8; ++j) {
      pb[j + 8 * hi]      = mrow[j];
      pb[16 + j + 8 * hi] = lrow[j];
    }
  }
  float* ab = pb + 32;
  #pragma unroll
  for (int j = 0; j < 8; ++j) {
    const int row = j + 8 * hi;
    ab[row * 64 + nlo]      = acc0[j];
    ab[row * 64 + 16 + nlo] = acc1[j];
    ab[row * 64 + 32 + nlo] = acc2[j];
    ab[row * 64 + 48 + nlo] = acc3[j];
  }
}

// ============================================================================
// attn_reduce: combine NCHUNK partials per head -> O[s, h*64+d]
// ============================================================================
__global__ __launch_bounds__(64) void attn_reduce_kernel(
    const float* __restrict__ part, float* __restrict__ O) {
  const int h = blockIdx.x;     // 16
  const int d = threadIdx.x;    // 64
  for (int s = 0; s < SEQ_S; ++s) {
    float m = -3.0e38f;
    for (int c = 0; c < NCHUNK; ++c)
      m = fmaxf(m, part[(size_t)(h * NCHUNK + c) * 1056 + s]);
    float l = 0.0f, a = 0.0f;
    for (int c = 0; c < NCHUNK; ++c) {
      const float* pb = part + (size_t)(h * NCHUNK + c) * 1056;
      const float f = __expf(pb[s] - m);
      l += pb[16 + s] * f;
      a += pb[32 + s * 64 + d] * f;
    }
    O[(size_t)s * D_MODEL + h * HEAD_DIM + d] = a / l;
  }
}

// ============================================================================
// launcher
// ============================================================================
extern "C" void kernel_launch(void* const* d_in, const int* in_sizes, int n_in,
                              void* d_out, int out_size, void* d_ws, size_t ws_size,
                              hipStream_t stream) {
  (void)in_sizes; (void)n_in; (void)out_size; (void)ws_size;
  const float* query  = (const float*)d_in[0];
  const float* Wq     = (const float*)d_in[1];
  const float* Wk     = (const float*)d_in[2];
  const float* Wv     = (const float*)d_in[3];
  const float* Wo     = (const float*)d_in[4];
  const float* cacheK = (const float*)d_in[5];
  const float* cacheV = (const float*)d_in[6];
  float* out = (float*)d_out;                 // [output | K | V], 3 x 16384 f32

  float* Qws  = (float*)d_ws;                 // 16384 f32
  float* Kws  = Qws + SEQ_S * D_MODEL;        // 16384 f32
  float* Vws  = Kws + SEQ_S * D_MODEL;        // 16384 f32
  float* Ows  = Vws + SEQ_S * D_MODEL;        // 16384 f32
  float* part = Ows + SEQ_S * D_MODEL;        // 512 * 1056 f32

  // Q/K/V projections (K,V mirrored straight into d_out as reference outputs)
  gemm16_kernel<<<D_MODEL / 16, 32, 0, stream>>>(query, Wq, Qws, nullptr);
  gemm16_kernel<<<D_MODEL / 16, 32, 0, stream>>>(query, Wk, Kws, out + 1 * SEQ_S * D_MODEL);
  gemm16_kernel<<<D_MODEL / 16, 32, 0, stream>>>(query, Wv, Vws, out + 2 * SEQ_S * D_MODEL);

  // flash-decode over 16 heads x 32 T-chunks
  attn_partial_kernel<<<N_HEADS * NCHUNK, 32, 0, stream>>>(
      Qws, Kws, Vws, cacheK, cacheV, part);
  attn_reduce_kernel<<<N_HEADS, 64, 0, stream>>>(part, Ows);

  // output projection
  gemm16_kernel<<<D_MODEL / 16, 32, 0, stream>>>(Ows, Wo, out, nullptr);
}